// TopKAttention_7344394076836
// MI455X (gfx1250) — compile-verified
//
#include <hip/hip_runtime.h>
#include <math.h>

typedef __attribute__((ext_vector_type(2))) float v2f;
typedef __attribute__((ext_vector_type(8))) float v8f;
typedef __attribute__((ext_vector_type(4))) unsigned int v4u;
typedef __attribute__((ext_vector_type(8))) int v8i;
typedef __attribute__((ext_vector_type(4))) int v4i;

// ---------------------------------------------------------------------------
// f32 WMMA helper: D = A(16x4) * B(4x16) + C(16x16), wave32.
// ---------------------------------------------------------------------------
__device__ __forceinline__ v8f wmma4(v2f a, v2f b, v8f c) {
  return __builtin_amdgcn_wmma_f32_16x16x4_f32(false, a, false, b, (short)0, c,
                                               false, false);
}

// A-fragment (and B-fragment from row-major [N,K]) loader for 16x16x4 f32:
// lane<16 holds [row=lane][k = 4*kk + {0,1}], lane>=16 holds [row=lane-16][4*kk+{2,3}]
__device__ __forceinline__ v2f ldfrag(const float* __restrict__ base, int ld,
                                      int kk) {
  const int lane = threadIdx.x & 31;
  const int row = lane & 15;
  const int kr = (kk << 2) + ((lane >> 4) << 1);
  return *(const v2f*)(base + (size_t)row * ld + kr);  // 8B aligned -> b64 load
}

// 16x64 output strip GEMM: C = A[16,K] * B_rows[64,K]^T, fp32, K%4==0
__device__ __forceinline__ void gemm_strip(const float* __restrict__ A, int lda,
                                           const float* __restrict__ B, int ldb,
                                           int K, v8f acc[4]) {
  for (int kk = 0; kk < (K >> 2); ++kk) {
    v2f a = ldfrag(A, lda, kk);
#pragma unroll
    for (int t = 0; t < 4; ++t) {
      v2f b = ldfrag(B + (size_t)(t << 4) * ldb, ldb, kk);
      acc[t] = wmma4(a, b, acc[t]);
    }
  }
}

// cross-lane xor reduction within 16-lane halves via ds_swizzle
template <int XM>
__device__ __forceinline__ float swz_xor(float x) {
  return __int_as_float(
      __builtin_amdgcn_ds_swizzle(__float_as_int(x), (XM << 10) | 0x1f));
}

// ---------------------------------------------------------------------------
// Problem constants
// ---------------------------------------------------------------------------
#define BB 4
#define LL 1024
#define NN 2048
#define VDIM 512
#define LDIM 768
#define HID 512
#define NH 8
#define HD 64
#define NBLK 32           // N / 64 column blocks in attention
#define MROW 65           // padded LDS row stride for mask tiles (floats)

// ---------------------------------------------------------------------------
// TDM: async DMA of one 16x64 f32 mask tile (row stride NN) into LDS.
// 2D descriptor, data_size=4B, pad 1 DWORD every 64 DWORDs -> LDS row = 65 f.
// ---------------------------------------------------------------------------
__device__ __forceinline__ void tdm_mask_tile(unsigned int lds_off,
                                              const float* g) {
  const unsigned long long ga = (unsigned long long)(uintptr_t)g;
  v4u g0 = {1u,                                   // count=1, user descriptor
            lds_off,                              // lds_addr (bytes)
            (unsigned int)ga,                     // global_addr[31:0]
            ((unsigned int)(ga >> 32) & 0x01FFFFFFu) | (2u << 30)};  // type=2
  v8i g1 = {(int)((2u << 16) | (1u << 20) | (5u << 22)),  // 4B | pad_en | every 64dw
            (int)(((unsigned int)NN & 0xFFFFu) << 16),    // tensor_dim0 lo16
            (int)(0x8000u << 16),                         // tensor_dim1 lo16 = 32768
            (int)(64u << 16),                             // tile_dim0 = 64
            16,                                           // tile_dim1 = 16
            (int)NN,                                      // tensor_dim0_stride
            0, 0};
  v4i gz4 = {0, 0, 0, 0};
  v8i gz8 = {0, 0, 0, 0, 0, 0, 0, 0};
  __builtin_amdgcn_tensor_load_to_lds(g0, g1, gz4, gz4, gz8, 0);
}

// ---------------------------------------------------------------------------
// Kernel 1: q = x @ Wq^T   -> stored [B,H,L,HD]
// ---------------------------------------------------------------------------
__global__ __launch_bounds__(32) void k_qproj(const float* __restrict__ x,
                                              const float* __restrict__ Wq,
                                              float* __restrict__ qout) {
  const int m0 = blockIdx.x << 4;  // M = B*L = 4096
  const int n0 = blockIdx.y << 6;  // N = HID = 512, 64-col strips
  v8f acc[4] = {};
  gemm_strip(x + (size_t)m0 * VDIM, VDIM, Wq + (size_t)n0 * VDIM, VDIM, VDIM, acc);
  const int lane = threadIdx.x & 31;
  const int col = lane & 15;
  const int ro = (lane >> 4) << 3;
#pragma unroll
  for (int t = 0; t < 4; ++t) {
    const int c = n0 + (t << 4) + col;
    const int h = c >> 6, d = c & 63;
#pragma unroll
    for (int r = 0; r < 8; ++r) {
      const int m = m0 + r + ro;
      const int b = m >> 10, l = m & (LL - 1);
      qout[(size_t)((((b << 3) + h) << 10) + l) * HD + d] = acc[t][r];
    }
  }
}

// ---------------------------------------------------------------------------
// Kernel 2: kv = vfeats @ Wkv^T -> k,v each stored [B,H,N,HD]
// ---------------------------------------------------------------------------
__global__ __launch_bounds__(32) void k_kvproj(const float* __restrict__ vf,
                                               const float* __restrict__ Wkv,
                                               float* __restrict__ kout,
                                               float* __restrict__ vout) {
  const int m0 = blockIdx.x << 4;  // M = B*N = 8192
  const int n0 = blockIdx.y << 6;  // N = 2*HID = 1024
  v8f acc[4] = {};
  gemm_strip(vf + (size_t)m0 * LDIM, LDIM, Wkv + (size_t)n0 * LDIM, LDIM, LDIM, acc);
  const int lane = threadIdx.x & 31;
  const int col = lane & 15;
  const int ro = (lane >> 4) << 3;
  float* __restrict__ dst = (n0 < HID) ? kout : vout;  // strip is head-aligned
#pragma unroll
  for (int t = 0; t < 4; ++t) {
    const int c = n0 + (t << 4) + col;
    const int h = (c & (HID - 1)) >> 6, d = c & 63;
#pragma unroll
    for (int r = 0; r < 8; ++r) {
      const int m = m0 + r + ro;
      const int b = m >> 11, n = m & (NN - 1);
      dst[(size_t)((((b << 3) + h) << 11) + n) * HD + d] = acc[t][r];
    }
  }
}

// ---------------------------------------------------------------------------
// Kernel 3: flash attention, 64-column N blocks, TDM-double-buffered mask.
// One wave32 per 16-row Q tile. Output stored [B,L,HID].
// ---------------------------------------------------------------------------
__global__ __launch_bounds__(32) void k_attn(const float* __restrict__ q,
                                             const float* __restrict__ kmat,
                                             const float* __restrict__ vmat,
                                             const float* __restrict__ mask,
                                             float* __restrict__ ao) {
  __shared__ float sM[2][16 * MROW];  // TDM mask tiles, double buffered
  __shared__ float sP[16][68];        // P (16x64) staging, padded rows
  const int tile = blockIdx.x;        // B*H*(L/16) = 2048
  const int lt = tile & 63;
  const int bh = tile >> 6;           // b*8+h
  const int h = bh & 7, b = bh >> 3;
  const int l0 = lt << 4;
  const float* __restrict__ qb = q + ((size_t)bh * LL + l0) * HD;
  const float* __restrict__ kb = kmat + (size_t)bh * NN * HD;
  const float* __restrict__ vb = vmat + (size_t)bh * NN * HD;
  const float* __restrict__ mb = mask + ((size_t)bh * LL + l0) * NN;
  const int lane = threadIdx.x & 31;
  const int col = lane & 15;
  const int hi2 = (lane >> 4) << 1;
  const int ro = (lane >> 4) << 3;

  // Q tile fragments kept in registers for the whole N sweep (32 VGPRs)
  v2f qa[16];
#pragma unroll
  for (int kk = 0; kk < 16; ++kk) qa[kk] = ldfrag(qb, HD, kk);

  v8f o[4] = {};
  float M[8], Lr[8];
#pragma unroll
  for (int r = 0; r < 8; ++r) { M[r] = -INFINITY; Lr[r] = 0.f; }

  const unsigned int ldsM0 = (unsigned int)(size_t)(void*)&sM[0][0];
  const unsigned int ldsM1 = (unsigned int)(size_t)(void*)&sM[1][0];

  tdm_mask_tile(ldsM0, mb);  // prime block 0

  for (int j = 0; j < NBLK; ++j) {
    // issue next mask tile into the other buffer, wait for current
    if (j + 1 < NBLK) {
      tdm_mask_tile((j & 1) ? ldsM0 : ldsM1, mb + (size_t)(j + 1) * 64);
      __builtin_amdgcn_s_wait_tensorcnt(1);
    } else {
      __builtin_amdgcn_s_wait_tensorcnt(0);
    }
    const float* __restrict__ smj = (j & 1) ? &sM[1][0] : &sM[0][0];

    // S[4] = Q * K_block^T   (64 WMMAs over HD=64, 4 column tiles)
    v8f s[4];
#pragma unroll
    for (int t = 0; t < 4; ++t) {
      const float* kt = kb + (size_t)((j << 2) + t) * 16 * HD;
      v8f acc = {};
#pragma unroll
      for (int kk = 0; kk < 16; ++kk) acc = wmma4(qa[kk], ldfrag(kt, HD, kk), acc);
      s[t] = acc;
    }

    if (j + 1 < NBLK)  // pull next V block toward the caches
      __builtin_prefetch(vb + (size_t)(j + 1) * 64 * HD, 0, 0);

    // scale + mask + online softmax over the 64-col block
    // (local max/sum over 4 tiles first, then one 4-step swizzle tree)
    float alpha[8];
#pragma unroll
    for (int r = 0; r < 8; ++r) {
      float sv[4];
      float mx = -INFINITY;
#pragma unroll
      for (int t = 0; t < 4; ++t) {
        const float mval = smj[(r + ro) * MROW + (t << 4) + col];
        sv[t] = s[t][r] * 0.125f + mval;  // SCALE = 64^-0.5
        mx = fmaxf(mx, sv[t]);
      }
      mx = fmaxf(mx, swz_xor<1>(mx));
      mx = fmaxf(mx, swz_xor<2>(mx));
      mx = fmaxf(mx, swz_xor<4>(mx));
      mx = fmaxf(mx, swz_xor<8>(mx));
      const float Mn = fmaxf(M[r], mx);
      alpha[r] = __expf(M[r] - Mn);
      M[r] = Mn;
      float rs = 0.f;
#pragma unroll
      for (int t = 0; t < 4; ++t) {
        const float pv = __expf(sv[t] - Mn);
        sP[r + ro][(t << 4) + col] = pv;
        rs += pv;
      }
      rs += swz_xor<1>(rs);
      rs += swz_xor<2>(rs);
      rs += swz_xor<4>(rs);
      rs += swz_xor<8>(rs);
      Lr[r] = Lr[r] * alpha[r] + rs;
    }
#pragma unroll
    for (int t = 0; t < 4; ++t)
#pragma unroll
      for (int r = 0; r < 8; ++r) o[t][r] *= alpha[r];
    __syncthreads();

    // O += P(16x64) * V_block(64x64)  (64 WMMAs)
    const float* vt = vb + (size_t)j * 64 * HD;
#pragma unroll
    for (int kk = 0; kk < 16; ++kk) {
      const int kr = (kk << 2) + hi2;
      const v2f pa = v2f{sP[col][kr], sP[col][kr + 1]};
#pragma unroll
      for (int t = 0; t < 4; ++t) {
        v2f bf;
        bf.x = vt[(size_t)kr * HD + (t << 4) + col];
        bf.y = vt[(size_t)(kr + 1) * HD + (t << 4) + col];
        o[t] = wmma4(pa, bf, o[t]);
      }
    }
    __syncthreads();
  }

  // normalize and scatter to [B,L,HID]
#pragma unroll
  for (int r = 0; r < 8; ++r) Lr[r] = 1.f / Lr[r];
#pragma unroll
  for (int t = 0; t < 4; ++t)
#pragma unroll
    for (int r = 0; r < 8; ++r) {
      const int l = l0 + r + ro;
      ao[((size_t)b * LL + l) * HID + (h << 6) + (t << 4) + col] =
          o[t][r] * Lr[r];
    }
}

// ---------------------------------------------------------------------------
// Kernel 4: out = attn_out @ Wproj^T + bproj
// ---------------------------------------------------------------------------
__global__ __launch_bounds__(32) void k_oproj(const float* __restrict__ ao,
                                              const float* __restrict__ Wp,
                                              const float* __restrict__ bp,
                                              float* __restrict__ out) {
  const int m0 = blockIdx.x << 4;  // M = 4096
  const int n0 = blockIdx.y << 6;  // N = 512
  v8f acc[4] = {};
  gemm_strip(ao + (size_t)m0 * HID, HID, Wp + (size_t)n0 * HID, HID, HID, acc);
  const int lane = threadIdx.x & 31;
  const int col = lane & 15;
  const int ro = (lane >> 4) << 3;
#pragma unroll
  for (int t = 0; t < 4; ++t) {
    const int c = n0 + (t << 4) + col;
    const float bias = bp[c];
#pragma unroll
    for (int r = 0; r < 8; ++r) {
      const int m = m0 + r + ro;
      out[(size_t)m * VDIM + c] = acc[t][r] + bias;
    }
  }
}

// ---------------------------------------------------------------------------
extern "C" void kernel_launch(void* const* d_in, const int* in_sizes, int n_in,
                              void* d_out, int out_size, void* d_ws,
                              size_t ws_size, hipStream_t stream) {
  const float* x    = (const float*)d_in[0];
  const float* vf   = (const float*)d_in[1];
  const float* mask = (const float*)d_in[2];
  const float* Wq   = (const float*)d_in[3];
  const float* Wkv  = (const float*)d_in[4];
  const float* Wp   = (const float*)d_in[5];
  const float* bp   = (const float*)d_in[6];
  float* out = (float*)d_out;

  float* q  = (float*)d_ws;                         // B*H*L*HD   = 2,097,152 f
  float* km = q + (size_t)BB * NH * LL * HD;        // B*H*N*HD   = 4,194,304 f
  float* vm = km + (size_t)BB * NH * NN * HD;       // B*H*N*HD   = 4,194,304 f
  float* ao = vm + (size_t)BB * NH * NN * HD;       // B*L*HID    = 2,097,152 f

  k_qproj<<<dim3((BB * LL) / 16, HID / 64), 32, 0, stream>>>(x, Wq, q);
  k_kvproj<<<dim3((BB * NN) / 16, (2 * HID) / 64), 32, 0, stream>>>(vf, Wkv, km, vm);
  k_attn<<<dim3(BB * NH * (LL / 16)), 32, 0, stream>>>(q, km, vm, mask, ao);
  k_oproj<<<dim3((BB * LL) / 16, VDIM / 64), 32, 0, stream>>>(ao, Wp, bp, out);
}